// CapsuleLayer_34316788695599
// MI455X (gfx1250) — compile-verified
//
#include <hip/hip_runtime.h>

// Problem constants (from reference)
#define BB   256
#define NC   10
#define NR   1152
#define INC  8
#define OC   16
#define PSTR 17   // padded LDS row stride (floats) to avoid bank conflicts

typedef float v2f __attribute__((ext_vector_type(2)));
typedef float v8f __attribute__((ext_vector_type(8)));

// ---------------------------------------------------------------------------
// Phase 1: priors[b,c,r,:] via V_WMMA_F32_16X16X4_F32 (fp32 WMMA).
// For fixed (c,r): P[b,o] = sum_i x[b,r,i] * W[c,r,i,o]  ->  [chunkB x 8] @ [8 x 16]
// One wave per (c,r); M = 16 batch rows per tile, K=8 split into two K=4 steps.
// A layout (16x4 f32): lanes 0-15 hold M=0..15; VGPR0 = K0 (lanes<16) / K2 (lanes>=16),
// VGPR1 = K1 / K3.  B (4x16) mirrors: lanes = N, K rows split across lane halves.
// ---------------------------------------------------------------------------
__global__ __launch_bounds__(256)
void caps_priors_wmma(const float* __restrict__ x,      // [B, NR, 8]
                      const float* __restrict__ w,      // [NC, NR, 8, 16]
                      float* __restrict__ priors,       // [chunkB, NC, NR, 16] (chunk-local)
                      int b0, int nbt)                  // nbt = chunkB/16
{
    const int wave   = threadIdx.x >> 5;
    const int lane   = threadIdx.x & 31;
    const int laneLo = lane & 15;
    const int kHalf  = lane >> 4;

    const int rg = blockIdx.x % (NR / 8);
    const int c  = blockIdx.x / (NR / 8);
    const int r  = rg * 8 + wave;

    // B operand: W[c,r,:,:] for both K-steps (loaded once, reused over batch tiles)
    const float* wr = w + ((size_t)c * NR + r) * (INC * OC);
    v2f Bk0, Bk1;
    {
        const int i0 = kHalf * 2;          // rows K = {0,1} or {2,3}
        Bk0.x = wr[(i0 + 0) * OC + laneLo];
        Bk0.y = wr[(i0 + 1) * OC + laneLo];
        const int i1 = 4 + kHalf * 2;      // rows K = {4,5} or {6,7}
        Bk1.x = wr[(i1 + 0) * OC + laneLo];
        Bk1.y = wr[(i1 + 1) * OC + laneLo];
    }

    const size_t mstride = (size_t)NC * NR * OC;   // stride between batch rows in priors

    for (int bt = 0; bt < nbt; ++bt) {
        const int b = b0 + bt * 16 + laneLo;       // M index = laneLo
        const float* xr = x + ((size_t)b * NR + r) * INC;
        v2f Ak0, Ak1;
        Ak0.x = xr[kHalf * 2 + 0];
        Ak0.y = xr[kHalf * 2 + 1];
        Ak1.x = xr[4 + kHalf * 2 + 0];
        Ak1.y = xr[4 + kHalf * 2 + 1];

        v8f acc = {};
        acc = __builtin_amdgcn_wmma_f32_16x16x4_f32(false, Ak0, false, Bk0,
                                                    (short)0, acc, false, false);
        acc = __builtin_amdgcn_wmma_f32_16x16x4_f32(false, Ak1, false, Bk1,
                                                    (short)0, acc, false, false);

        // D layout: VGPR j holds M = kHalf*8 + j, N = laneLo
        float* pout = priors +
            (((size_t)(bt * 16 + kHalf * 8) * NC + c) * NR + r) * OC + laneLo;
        #pragma unroll
        for (int j = 0; j < 8; ++j)
            pout[(size_t)j * mstride] = acc[j];
    }
}

// ---------------------------------------------------------------------------
// Phase 2: dynamic routing, one 256-thread block per (b, c).
// priors tile (73.7 KB) held in LDS with padded rows; 3 softmax/squash iters.
// ---------------------------------------------------------------------------
__global__ __launch_bounds__(256)
void caps_route(const float* __restrict__ priors,  // [chunkB, NC, NR, OC]
                float* __restrict__ out,           // [B, NC, OC]
                int b0)
{
    __shared__ float P[NR * PSTR];   // 78336 B, padded priors tile
    __shared__ float logits[NR];
    __shared__ float ev[NR];
    __shared__ float red[256];
    __shared__ float sv[OC];
    __shared__ float vv[OC];

    const int t  = threadIdx.x;
    const int lb = blockIdx.x / NC;
    const int c  = blockIdx.x % NC;

    // Load priors tile into LDS (float4 global reads, scalar padded LDS writes)
    const float* g = priors + ((size_t)lb * NC + c) * (NR * OC);
    for (int i = t; i < (NR * OC) / 4; i += 256) {
        const float4 v4 = ((const float4*)g)[i];
        const int gi = i * 4;
        const int rr = gi >> 4;
        const int oo = gi & 15;
        float* dst = &P[rr * PSTR + oo];
        dst[0] = v4.x; dst[1] = v4.y; dst[2] = v4.z; dst[3] = v4.w;
    }
    for (int r = t; r < NR; r += 256) logits[r] = 0.f;
    __syncthreads();

    const int o  = t >> 4;   // 0..15 : output channel handled in s-reduction
    const int ro = t & 15;   // r offset

    for (int it = 0; it < 3; ++it) {
        // ---- softmax over route axis: max ----
        float lmax = -3.402823466e38f;
        for (int r = t; r < NR; r += 256) lmax = fmaxf(lmax, logits[r]);
        red[t] = lmax; __syncthreads();
        for (int s = 128; s > 0; s >>= 1) {
            if (t < s) red[t] = fmaxf(red[t], red[t + s]);
            __syncthreads();
        }
        const float mx = red[0];
        __syncthreads();

        // ---- exp + denom ----
        float lsum = 0.f;
        for (int r = t; r < NR; r += 256) {
            const float e = __expf(logits[r] - mx);
            ev[r] = e; lsum += e;
        }
        red[t] = lsum; __syncthreads();
        for (int s = 128; s > 0; s >>= 1) {
            if (t < s) red[t] += red[t + s];
            __syncthreads();
        }
        const float invden = 1.f / red[0];
        __syncthreads();

        // ---- s[o] = invden * sum_r ev[r] * P[r][o] ----
        float acc = 0.f;
        for (int k = 0; k < NR / 16; ++k) {
            const int r = ro + 16 * k;
            acc += ev[r] * P[r * PSTR + o];
        }
        red[t] = acc; __syncthreads();
        if (t < OC) {
            float ssum = 0.f;
            #pragma unroll
            for (int j = 0; j < 16; ++j) ssum += red[t * 16 + j];
            sv[t] = ssum * invden;
        }
        __syncthreads();

        // ---- squash: v = s * sqrt(sq) / (1 + sq),  sq = |s|^2 ----
        if (t == 0) {
            float sq = 0.f;
            #pragma unroll
            for (int j = 0; j < OC; ++j) sq += sv[j] * sv[j];
            const float scale = __fsqrt_rn(sq) / (1.f + sq);
            #pragma unroll
            for (int j = 0; j < OC; ++j) vv[j] = sv[j] * scale;
        }
        __syncthreads();

        if (it < 2) {
            // logits[r] += P[r,:] . v
            for (int r = t; r < NR; r += 256) {
                float d = 0.f;
                #pragma unroll
                for (int j = 0; j < OC; ++j) d += P[r * PSTR + j] * vv[j];
                logits[r] += d;
            }
            __syncthreads();
        }
    }

    if (t < OC)
        out[(((size_t)(b0 + lb)) * NC + c) * OC + t] = vv[t];
}

// ---------------------------------------------------------------------------
extern "C" void kernel_launch(void* const* d_in, const int* in_sizes, int n_in,
                              void* d_out, int out_size, void* d_ws, size_t ws_size,
                              hipStream_t stream) {
    (void)in_sizes; (void)n_in; (void)out_size;
    const float* x = (const float*)d_in[0];
    const float* w = (const float*)d_in[1];
    float* out     = (float*)d_out;
    float* priors  = (float*)d_ws;

    // Batch chunking: keep the priors tile (chunkB * 737280 B) L2-resident.
    const size_t perB = (size_t)NC * NR * OC * sizeof(float);  // 737280 B
    int chunkB = 32;                                           // 23.6 MB << 192 MB L2
    while (chunkB > 16 && (size_t)chunkB * perB > ws_size) chunkB >>= 1;

    for (int b0 = 0; b0 < BB; b0 += chunkB) {
        caps_priors_wmma<<<dim3(NC * (NR / 8)), dim3(256), 0, stream>>>(
            x, w, priors, b0, chunkB / 16);
        caps_route<<<dim3(chunkB * NC), dim3(256), 0, stream>>>(priors, out, b0);
    }
}